// TnoFFT_2508260900873
// MI455X (gfx1250) — compile-verified
//
#include <hip/hip_runtime.h>

// CDNA5 / gfx1250: fp32 Toeplitz-GEMM formulation of the FFT conv reference.
// o[b,j,d] = sum_i x[b,i,d] * t[(j-i) mod 8192, d]
// One workgroup per channel d; V_WMMA_F32_16X16X4_F32 with 16 B-columns =
// 4 batches x 4 row-shifts (Toeplitz shift identity) for full N utilization.
// Reversed-t table is stored UNWRAPPED (periodic extension) so inner-loop LDS
// addresses are pure linear induction variables (no per-iteration AND).

typedef float v2f __attribute__((ext_vector_type(2)));
typedef float v8f __attribute__((ext_vector_type(8)));

#define BB   4
#define NN   4096
#define DD   1024
#define LL   8192            // 2*N, circular period
#define PADL 48              // max shift (3*16)
#define KHAT 4144            // N + PADL, contraction length (multiple of 4)
#define XSTR 4208            // padded x stride in dwords (bank-decorrelated)
#define TEXT 12336           // extended reversed-t table: 8191 + 4140 + 2, rounded

__global__ __launch_bounds__(512, 1)
void toeplitz_wmma_f32_kernel(const float* __restrict__ x,
                              const float* __restrict__ t,
                              float* __restrict__ out) {
    __shared__ float trs[TEXT];          // reversed t, periodically extended
    __shared__ float xsh[BB * XSTR];     // zero-padded x columns

    const int d   = blockIdx.x;
    const int tid = threadIdx.x;

    // ---- stage reversed t column: trs[q] = t[(PADL - q) mod L, d] ----
    for (int q = tid; q < TEXT; q += 512) {
        int row = (PADL - q) & (LL - 1);
        trs[q] = t[row * DD + d];
    }
    // ---- stage x columns, 48 leading zeros + tail zeros ----
    for (int bq = tid; bq < BB * XSTR; bq += 512) {
        int b = bq / XSTR;
        int q = bq - b * XSTR;
        int i = q - PADL;
        float v = 0.0f;
        if (i >= 0 && i < NN) v = x[(b * NN + i) * DD + d];
        xsh[bq] = v;
    }
    __syncthreads();

    const int lane = tid & 31;
    const int wave = tid >> 5;           // 16 waves
    const int m    = lane & 15;          // A row / D column index
    const int g    = lane >> 4;          // lane half: K sub-group
    const int p    = m >> 2;             // shift index (delta = 16*p)
    const int bcol = m & 3;              // batch index

    // B operand base: xsh[bcol][k + 2g + 16p]
    const float* xb = &xsh[bcol * XSTR + 16 * p + 2 * g];

    // 4 j-tiles per wave: j0_w = 64*wave + 1024*w
    const int j0 = 64 * wave;
    v8f acc[4];
    const float* ta[4];
#pragma unroll
    for (int w = 0; w < 4; ++w) {
        acc[w] = (v8f){};
        // A index base: q = (2g - j0_w - m) mod L; k advances linearly (no wrap,
        // trs is periodically extended to TEXT entries).
        int qw = (2 * g - (j0 + 1024 * w) - m) & (LL - 1);
        ta[w] = &trs[qw];
    }

#pragma unroll 4
    for (int k = 0; k < KHAT; k += 4) {
        v2f Bv;
        Bv.x = xb[k];
        Bv.y = xb[k + 1];
#pragma unroll
        for (int w = 0; w < 4; ++w) {
            v2f Av;
            Av.x = ta[w][k];         // A[m, 2g]   = t[(j0w+m+48-(k+2g)) mod L]
            Av.y = ta[w][k + 1];     // A[m, 2g+1]
            acc[w] = __builtin_amdgcn_wmma_f32_16x16x4_f32(
                false, Av, false, Bv, (short)0, acc[w], false, false);
        }
    }

    // ---- store: D[r,n] -> out[bcol, j0_w + 16p + r, d]; r = v + 8*g ----
#pragma unroll
    for (int w = 0; w < 4; ++w) {
        int jbase = j0 + 1024 * w + 16 * p;
#pragma unroll
        for (int v = 0; v < 8; ++v) {
            int r = v + 8 * g;
            out[(bcol * NN + jbase + r) * DD + d] = acc[w][v];
        }
    }
}

extern "C" void kernel_launch(void* const* d_in, const int* in_sizes, int n_in,
                              void* d_out, int out_size, void* d_ws, size_t ws_size,
                              hipStream_t stream) {
    const float* x = (const float*)d_in[0];
    const float* t = (const float*)d_in[1];
    float* out = (float*)d_out;
    (void)in_sizes; (void)n_in; (void)out_size; (void)d_ws; (void)ws_size;

    dim3 grid(DD);    // one workgroup per channel d
    dim3 block(512);  // 16 waves (wave32)
    hipLaunchKernelGGL(toeplitz_wmma_f32_kernel, grid, block, 0, stream,
                       x, t, out);
}